// BilateralAntiAlias_25881472926180
// MI455X (gfx1250) — compile-verified
//
#include <hip/hip_runtime.h>
#include <cstdint>
#include <cstddef>

// Bilateral 5x5 filter, (16,3,512,512) f32.
// Compute-bound (~140 VALU + 25 trans per pixel vs ~8 B/pixel traffic), so:
//  - fold spatial weight into exponent: w = exp2(fma(d^2, -50/ln2, log2(s)))
//  - stage halo tiles into LDS with CDNA5 async-to-LDS DMA (ASYNCcnt path)
//  - normalize once per pixel with v_rcp_f32

#define KS      5
#define PADR    2
#define TILE    32
#define LW      (TILE + 2 * PADR)   // 36
#define LSTRIDE 37                  // odd stride -> conflict-free column access
#define IMH     512
#define IMW     512

__device__ __forceinline__ int reflect512(int i) {
    // jnp.pad(..., mode='reflect'): -1->1, -2->2, 512->510, 513->509 (pad<=2: one fold)
    i = (i < 0) ? -i : i;
    i = (i >= IMW) ? (2 * IMW - 2 - i) : i;
    return i;
}

__global__ void __launch_bounds__(256)
bilateral55_kernel(const float* __restrict__ img,
                   const float* __restrict__ spatial,
                   float* __restrict__ out)
{
    __shared__ float tile[LW * LSTRIDE];
    __shared__ float ls[KS * KS];

    const int tid     = threadIdx.x;
    const int bc      = blockIdx.y;                      // b*C + c
    const int tile_id = blockIdx.x;                      // 0..255
    const int ty0     = (tile_id / (IMW / TILE)) * TILE;
    const int tx0     = (tile_id % (IMW / TILE)) * TILE;

    const float* __restrict__ chan = img + (size_t)bc * IMH * IMW;

    // Per-block: log2 of the 25 spatial weights (folds the multiply into exp2).
    if (tid < KS * KS) {
        // __builtin_amdgcn_logf is base-2 log -> v_log_f32
        ls[tid] = __builtin_amdgcn_logf(spatial[tid]);
    }

    // ---- Stage 36x36 halo tile into LDS via CDNA5 async-to-LDS DMA ----
    for (int idx = tid; idx < LW * LW; idx += 256) {
        const int ly = idx / LW;
        const int lx = idx - ly * LW;
        const int gy = reflect512(ty0 - PADR + ly);
        const int gx = reflect512(tx0 - PADR + lx);
        const float* gptr = chan + (size_t)gy * IMW + gx;
        // Low 32 bits of a generic LDS pointer are the LDS byte offset.
        const uint32_t lds_off = (uint32_t)(uintptr_t)&tile[ly * LSTRIDE + lx];
        asm volatile("global_load_async_to_lds_b32 %0, %1, off"
                     :: "v"(lds_off), "v"(gptr)
                     : "memory");
    }
    // Drain this wave's ASYNCcnt, then make the tile visible workgroup-wide.
    asm volatile("s_wait_asynccnt 0" ::: "memory");
    __syncthreads();

    // ---- Compute: each thread owns 4 consecutive rows of a 32-wide strip ----
    const int tx = tid & 31;        // lane -> x (coalesced global stores)
    const int ty = tid >> 5;        // 0..7
    const float NC = -72.13475204444817f;   // -50 / ln(2)  (sigma_int = 0.1)

#pragma unroll
    for (int r = 0; r < 4; ++r) {
        const int oy = ty * 4 + r;
        const float* row0 = &tile[oy * LSTRIDE + tx];
        const float center = row0[2 * LSTRIDE + 2];
        float wsum = 0.0f;
        float acc  = 0.0f;
#pragma unroll
        for (int ky = 0; ky < KS; ++ky) {
#pragma unroll
            for (int kx = 0; kx < KS; ++kx) {
                const float v = row0[ky * LSTRIDE + kx];
                const float d = v - center;
                // w = spatial * exp(-d^2/(2*0.1^2)) == exp2(fma(d^2, NC, log2(spatial)))
                const float w = __builtin_amdgcn_exp2f(fmaf(d * d, NC, ls[ky * KS + kx]));
                wsum += w;
                acc = fmaf(w, v, acc);
            }
        }
        // filtered = sum(w*p) / (sum(w) + eps)   (identical to normalize-then-sum)
        const float res = acc * __builtin_amdgcn_rcpf(wsum + 1e-8f);
        out[(size_t)bc * IMH * IMW + (size_t)(ty0 + oy) * IMW + (tx0 + tx)] = res;
    }
}

extern "C" void kernel_launch(void* const* d_in, const int* in_sizes, int n_in,
                              void* d_out, int out_size, void* d_ws, size_t ws_size,
                              hipStream_t stream) {
    const float* img     = (const float*)d_in[0];   // (16,3,512,512) f32
    const float* spatial = (const float*)d_in[1];   // (5,5) f32
    float* out           = (float*)d_out;

    const int bc_total = in_sizes[0] / (IMH * IMW); // 48
    dim3 grid((IMH / TILE) * (IMW / TILE), bc_total);
    bilateral55_kernel<<<grid, 256, 0, stream>>>(img, spatial, out);
}